// Model_80461917323557
// MI455X (gfx1250) — compile-verified
//
#include <hip/hip_runtime.h>

// ---------------- problem constants ----------------
#define B_   2
#define T_   12
#define N_   20000
#define D_   64
#define SH_  2
#define TH_  6
#define HD_  8
#define E_   160000
#define BT_  (B_ * T_)          // 24
#define R_   (B_ * T_ * N_)     // 480000 rows of x flattened over (b,t,n)

typedef __attribute__((ext_vector_type(2))) float v2f;
typedef __attribute__((ext_vector_type(8))) float v8f;

// D = A(16x4) * B(4x16) + C  in fp32 on the matrix pipe (CDNA5 V_WMMA_F32_16X16X4_F32)
__device__ __forceinline__ v8f wmma4(v2f a, v2f b, v8f c) {
    return __builtin_amdgcn_wmma_f32_16x16x4_f32(false, a, false, b, (short)0, c,
                                                 false, false);
}

// =====================================================================
// Kernel 1: GAT node prep.  h = x @ W_gat (64 -> 16), s_s = <h, a_src>,
// s_d = <h, a_dst> per (sh).  One wave handles a 16-row tile of the
// 480000 flattened rows (tiles never straddle a (b,t) since N%16==0).
// =====================================================================
__global__ __launch_bounds__(128) void gat_prep(
    const float* __restrict__ x,       // [B,T,N,64]
    const float* __restrict__ Wg,      // [64,16] row-major (d,k)
    const float* __restrict__ a_src,   // [2,8]
    const float* __restrict__ a_dst,   // [2,8]
    float* __restrict__ h_buf,         // [R,16]
    float* __restrict__ ss_buf,        // [R,2]
    float* __restrict__ sd_buf)        // [R,2]
{
    __shared__ __align__(16) float Wg_s[64 * 16];
    __shared__ float as_s[16], ad_s[16];
    __shared__ __align__(16) float Xa[4][16 * 64];
    __shared__ __align__(16) float Ht[4][16 * 16];

    const int tid = threadIdx.x;
    for (int i = tid; i < 64 * 16; i += 128) Wg_s[i] = Wg[i];
    if (tid < 16) { as_s[tid] = a_src[tid]; ad_s[tid] = a_dst[tid]; }
    __syncthreads();

    const int wave = tid >> 5, lane = tid & 31;
    const int hi   = lane >> 4;         // 0/1 half of the wave
    const int mrow = lane & 15;         // row for A-frag, col for B/C-frag
    const int kb0  = hi * 2;

    const long tile = (long)blockIdx.x * 4 + wave;   // 30000 tiles
    const long row0 = tile * 16;

    // 16x64 contiguous row tile -> LDS (float4)
    {
        const float4* xs = (const float4*)(x + row0 * 64);
        float4* xd = (float4*)Xa[wave];
        for (int i = lane; i < 16 * 16; i += 32) xd[i] = xs[i];
    }

    v8f c = {};
    for (int ks = 0; ks < 16; ++ks) {          // K = 64 in steps of 4
        const int kb = ks * 4 + kb0;
        v2f a, b;
        a.x = Xa[wave][mrow * 64 + kb];
        a.y = Xa[wave][mrow * 64 + kb + 1];
        b.x = Wg_s[kb * 16 + mrow];            // B[d][k] = W_gat[d][k]
        b.y = Wg_s[(kb + 1) * 16 + mrow];
        c = wmma4(a, b, c);
    }

    // write h (global) and stash in LDS for the s reductions
    #pragma unroll
    for (int r = 0; r < 8; ++r) {
        const int row = r + 8 * hi;
        h_buf[(row0 + row) * 16 + mrow] = c[r];
        Ht[wave][row * 16 + mrow] = c[r];
    }

    if (lane < 16) {
        const int row = lane;
        float s0 = 0.f, s1 = 0.f, d0 = 0.f, d1 = 0.f;
        #pragma unroll
        for (int hd = 0; hd < 8; ++hd) {
            const float h0 = Ht[wave][row * 16 + hd];
            const float h1 = Ht[wave][row * 16 + 8 + hd];
            s0 += h0 * as_s[hd];     s1 += h1 * as_s[8 + hd];
            d0 += h0 * ad_s[hd];     d1 += h1 * ad_s[8 + hd];
        }
        ss_buf[(row0 + row) * 2 + 0] = s0;
        ss_buf[(row0 + row) * 2 + 1] = s1;
        sd_buf[(row0 + row) * 2 + 0] = d0;
        sd_buf[(row0 + row) * 2 + 1] = d1;
    }
}

// =====================================================================
// Kernel 2: edge scatter.  alpha = ex/den cancels the segment max, and
// segment_sum(alpha*h) = segment_sum(ex*h)/den, so one pass accumulates
// num[dst] (16f) and den[dst] (2f) with f32 atomics; divide happens in
// kernel 3.  Accumulators are [n][bt][*] so same-dst atomics share L2
// lines; the whole 35 MB accumulator lives in the 192 MB L2.
// =====================================================================
__global__ __launch_bounds__(256) void gat_edges(
    const int*   __restrict__ ei,      // [2,E]
    const float* __restrict__ ep,      // [E]
    const float* __restrict__ h_buf,   // [R,16]
    const float* __restrict__ ss_buf,  // [R,2]
    const float* __restrict__ sd_buf,  // [R,2]
    float* __restrict__ num,           // [N,24,16]
    float* __restrict__ den)           // [N,24,2]
{
    const int e  = blockIdx.x * 256 + threadIdx.x;
    const int bt = blockIdx.y;                     // 0..23, uniform per block
    if (e >= E_) return;

    const int src = ei[e];
    const int dst = ei[E_ + e];
    const float p = ep[e];
    const long btrow = (long)bt * N_;

    float hv[16];
    {
        const float4* hs = (const float4*)(h_buf + (btrow + src) * 16);
        #pragma unroll
        for (int i = 0; i < 4; ++i) ((float4*)hv)[i] = hs[i];
    }

    float* nm = num + ((long)dst * BT_ + bt) * 16;
    float* dn = den + ((long)dst * BT_ + bt) * 2;

    #pragma unroll
    for (int sh = 0; sh < 2; ++sh) {
        const float es = ss_buf[(btrow + src) * 2 + sh] +
                         sd_buf[(btrow + dst) * 2 + sh];
        const float ev = es > 0.f ? es : 0.2f * es;     // leaky_relu(0.2)
        const float w  = __expf(ev) * p;
        atomicAdd(dn + sh, w);
        #pragma unroll
        for (int hd = 0; hd < 8; ++hd)
            atomicAdd(nm + sh * 8 + hd, w * hv[sh * 8 + hd]);
    }
}

// =====================================================================
// Kernel 3: per-(b,n) temporal attention + concat + output projection.
// One wave per (b,n); 4 waves per block share the fp32 weights in LDS.
// All matmuls run on v_wmma_f32_16x16x4_f32 with T=12 padded to M=16.
// =====================================================================
__global__ __launch_bounds__(128) void temporal_out(
    const float* __restrict__ x,    // [B,T,N,64]
    const float* __restrict__ Wq,   // [48,64]
    const float* __restrict__ Wk,
    const float* __restrict__ Wv,
    const float* __restrict__ Wp,   // [64,64]
    const float* __restrict__ bp,   // [64]
    const float* __restrict__ num,  // [N,24,16]
    const float* __restrict__ den,  // [N,24,2]
    float* __restrict__ out)        // [B,T,N,64]
{
    __shared__ __align__(16) float Wqkv_s[3 * 48 * 64];   // q|k|v, [o][d]
    __shared__ __align__(16) float Wp_s[64 * 64];
    __shared__ float bp_s[64];
    __shared__ __align__(16) float CatX[4][16 * 64];      // X during QKV, cat after
    __shared__ __align__(16) float QKV[4][16 * 144];      // per row: q(48)|k(48)|v(48)
    __shared__ __align__(16) float Sc[4][16 * 16];        // scores / probs

    const int tid = threadIdx.x;
    for (int i = tid; i < 3 * 48 * 64; i += 128)
        Wqkv_s[i] = (i < 3072) ? Wq[i] : (i < 6144 ? Wk[i - 3072] : Wv[i - 6144]);
    for (int i = tid; i < 64 * 64; i += 128) Wp_s[i] = Wp[i];
    if (tid < 64) bp_s[tid] = bp[tid];
    __syncthreads();

    const int wave = tid >> 5, lane = tid & 31;
    const int hi   = lane >> 4;
    const int mrow = lane & 15;
    const int kb0  = hi * 2;

    const int pair = blockIdx.x * 4 + wave;     // 0..B*N-1
    const int b = pair / N_;
    const int n = pair % N_;

    float* Xl  = CatX[wave];
    float* qkv = QKV[wave];
    float* sc  = Sc[wave];

    // ---- stage x[b, :, n, :] (12x64, pad to 16) into LDS ----
    {
        const float* xb = x + ((long)b * T_ * N_ + n) * 64;
        for (int i = lane; i < 16 * 16; i += 32) {
            const int row = i >> 4, d4 = i & 15;
            float4 v = make_float4(0.f, 0.f, 0.f, 0.f);
            if (row < T_) v = *(const float4*)(xb + (long)row * N_ * 64 + d4 * 4);
            *(float4*)(Xl + row * 64 + d4 * 4) = v;
        }
    }

    // ---- QKV projections: 16x64 @ 64x48 each, 3 N-tiles, 16 k-steps ----
    for (int p = 0; p < 3; ++p) {
        v8f c0 = {}, c1 = {}, c2 = {};
        const float* W = Wqkv_s + p * 3072;     // [o][d]
        for (int ks = 0; ks < 16; ++ks) {
            const int kb = ks * 4 + kb0;
            v2f a, b0, b1, b2;
            a.x  = Xl[mrow * 64 + kb];
            a.y  = Xl[mrow * 64 + kb + 1];
            b0.x = W[mrow * 64 + kb];          b0.y = W[mrow * 64 + kb + 1];
            b1.x = W[(mrow + 16) * 64 + kb];   b1.y = W[(mrow + 16) * 64 + kb + 1];
            b2.x = W[(mrow + 32) * 64 + kb];   b2.y = W[(mrow + 32) * 64 + kb + 1];
            c0 = wmma4(a, b0, c0);
            c1 = wmma4(a, b1, c1);
            c2 = wmma4(a, b2, c2);
        }
        #pragma unroll
        for (int r = 0; r < 8; ++r) {
            const int row = r + 8 * hi;
            qkv[row * 144 + p * 48 + mrow]      = c0[r];
            qkv[row * 144 + p * 48 + 16 + mrow] = c1[r];
            qkv[row * 144 + p * 48 + 32 + mrow] = c2[r];
        }
    }

    // ---- attention per head: scores -> softmax -> P@V -> cat ----
    const float scale = 0.35355339059327373f;   // HD^-0.5
    for (int h = 0; h < TH_; ++h) {
        v8f s = {};
        for (int ks = 0; ks < 2; ++ks) {        // K = 8
            const int kk = ks * 4 + kb0;
            v2f a, bb;
            a.x  = qkv[mrow * 144 + h * 8 + kk];          // q[row][hd]
            a.y  = qkv[mrow * 144 + h * 8 + kk + 1];
            bb.x = qkv[mrow * 144 + 48 + h * 8 + kk];     // k[col][hd]
            bb.y = qkv[mrow * 144 + 48 + h * 8 + kk + 1];
            s = wmma4(a, bb, s);
        }
        #pragma unroll
        for (int r = 0; r < 8; ++r)
            sc[(r + 8 * hi) * 16 + mrow] = s[r] * scale;

        if (lane < 16) {                         // softmax per row (T=12)
            const int row = lane;
            if (row < T_) {
                float m = -1e30f;
                for (int j = 0; j < T_; ++j) m = fmaxf(m, sc[row * 16 + j]);
                float pb[T_]; float sum = 0.f;
                for (int j = 0; j < T_; ++j) {
                    const float e = __expf(sc[row * 16 + j] - m);
                    pb[j] = e; sum += e;
                }
                const float inv = 1.f / sum;
                for (int j = 0; j < T_; ++j) sc[row * 16 + j] = pb[j] * inv;
                for (int j = T_; j < 16; ++j)  sc[row * 16 + j] = 0.f;
            } else {
                for (int j = 0; j < 16; ++j)   sc[row * 16 + j] = 0.f;
            }
        }

        v8f th = {};
        for (int ks = 0; ks < 4; ++ks) {         // K = 16 (padded rows are 0)
            const int kk = ks * 4 + kb0;
            v2f a, bb;
            a.x = sc[mrow * 16 + kk];
            a.y = sc[mrow * 16 + kk + 1];
            const bool valid = mrow < 8;         // N cols 8..15 unused
            bb.x = valid ? qkv[kk * 144 + 96 + h * 8 + mrow] : 0.f;
            bb.y = valid ? qkv[(kk + 1) * 144 + 96 + h * 8 + mrow] : 0.f;
            th = wmma4(a, bb, th);
        }
        if (mrow < 8) {                          // cat[:, h*8 + mrow] = t_head
            #pragma unroll
            for (int r = 0; r < 8; ++r)
                Xl[(r + 8 * hi) * 64 + h * 8 + mrow] = th[r];
        }
    }

    // ---- s_x = num / max(den, 1e-16) into cat cols 48..63 ----
    {
        const float* nmb = num + ((long)n * BT_ + b * T_) * 16;
        const float* dnb = den + ((long)n * BT_ + b * T_) * 2;
        for (int i = lane; i < 16 * 16; i += 32) {
            const int row = i >> 4, cc = i & 15;
            float v = 0.f;
            if (row < T_) {
                const float d = dnb[row * 2 + (cc >> 3)];
                v = nmb[row * 16 + cc] / fmaxf(d, 1e-16f);
            }
            Xl[row * 64 + 48 + cc] = v;
        }
    }

    // ---- out = cat @ Wp^T + bp  (16x64 @ 64x64) ----
    v8f oacc[4] = {{}, {}, {}, {}};
    for (int ks = 0; ks < 16; ++ks) {
        const int kb = ks * 4 + kb0;
        v2f a;
        a.x = Xl[mrow * 64 + kb];
        a.y = Xl[mrow * 64 + kb + 1];
        #pragma unroll
        for (int nt = 0; nt < 4; ++nt) {
            v2f bb;
            bb.x = Wp_s[(nt * 16 + mrow) * 64 + kb];
            bb.y = Wp_s[(nt * 16 + mrow) * 64 + kb + 1];
            oacc[nt] = wmma4(a, bb, oacc[nt]);
        }
    }
    float* ob = out + ((long)b * T_ * N_ + n) * 64;
    #pragma unroll
    for (int nt = 0; nt < 4; ++nt) {
        const int o = nt * 16 + mrow;
        const float bias = bp_s[o];
        #pragma unroll
        for (int r = 0; r < 8; ++r) {
            const int row = r + 8 * hi;
            if (row < T_) ob[(long)row * N_ * 64 + o] = oacc[nt][r] + bias;
        }
    }
}

// =====================================================================
// launch:  ws layout (floats): h[R*16] | ss[R*2] | sd[R*2] | num[N*24*16]
// | den[N*24*2]  -> 18.24M floats = 73 MB total scratch.
// =====================================================================
extern "C" void kernel_launch(void* const* d_in, const int* in_sizes, int n_in,
                              void* d_out, int out_size, void* d_ws, size_t ws_size,
                              hipStream_t stream)
{
    const float* x   = (const float*)d_in[0];
    const float* Wq  = (const float*)d_in[1];
    const float* Wk  = (const float*)d_in[2];
    const float* Wv  = (const float*)d_in[3];
    const float* Wg  = (const float*)d_in[4];
    const float* asr = (const float*)d_in[5];
    const float* ads = (const float*)d_in[6];
    const float* Wp  = (const float*)d_in[7];
    const float* bp  = (const float*)d_in[8];
    const int*   ei  = (const int*)d_in[9];
    const float* ep  = (const float*)d_in[10];
    float* out = (float*)d_out;

    float* ws     = (float*)d_ws;
    float* h_buf  = ws;
    float* ss_buf = h_buf + (size_t)R_ * 16;
    float* sd_buf = ss_buf + (size_t)R_ * 2;
    float* num    = sd_buf + (size_t)R_ * 2;
    float* den    = num + (size_t)N_ * BT_ * 16;

    // zero the atomic accumulators every call (harness does not re-poison)
    hipMemsetAsync(num, 0, (size_t)(N_ * BT_ * 16 + N_ * BT_ * 2) * sizeof(float),
                   stream);

    gat_prep<<<R_ / 64, 128, 0, stream>>>(x, Wg, asr, ads, h_buf, ss_buf, sd_buf);
    gat_edges<<<dim3((E_ + 255) / 256, BT_), 256, 0, stream>>>(
        ei, ep, h_buf, ss_buf, sd_buf, num, den);
    temporal_out<<<(B_ * N_) / 4, 128, 0, stream>>>(
        x, Wq, Wk, Wv, Wp, bp, num, den, out);
}